// calculate_IoU_14482629722430
// MI455X (gfx1250) — compile-verified
//
#include <hip/hip_runtime.h>

typedef __attribute__((ext_vector_type(4))) int   v4i;
typedef __attribute__((ext_vector_type(2))) float v2f;
typedef __attribute__((ext_vector_type(8))) float v8f;

#define NCLS   21       // pred.shape[1] in the reference
#define IGN    255
#define NSAMP  8
#define BLK    256      // 8 waves of 32
#define GRIDX  512

// ---- per-element predicate accumulation --------------------------------
// valid = (g != 255). Since NCLS-1 < 255, g in [1,NCLS) implies valid, so:
//   tp  = (g==p) & g_in
//   fn  = g_in_count - tp   (accumulated implicitly)
//   fp  = (g!=p) & p_in & (g != 255)
__device__ __forceinline__ void acc_elem(int g, int p,
                                         unsigned& tp, unsigned& fp, unsigned& gin_cnt) {
  bool eq  = (g == p);
  bool gin = ((unsigned)(g - 1)) < (unsigned)(NCLS - 1);
  bool pin = ((unsigned)(p - 1)) < (unsigned)(NCLS - 1);
  bool vld = (g != IGN);
  gin_cnt += (unsigned)gin;
  tp      += (unsigned)(eq & gin);
  fp      += (unsigned)((!eq) & pin & vld);
}

// ---- wave32 sum via V_WMMA_F32_16X16X4_F32 -----------------------------
// A(16x4): VGPR0 = {lanes0-15: K=0, lanes16-31: K=2}, VGPR1 = 0.
// B(4x16) = all ones  =>  D[m][n] = x[lane m] + x[lane m+16].
// D VGPR v: lanes0-15 hold M=v, lanes16-31 hold M=v+8 -> sum the 8 D regs
// locally, then one xor-16 shuffle adds the two row halves.
__device__ __forceinline__ float wave_sum_wmma(float x) {
  v2f a; a[0] = x;    a[1] = 0.0f;
  v2f b; b[0] = 1.0f; b[1] = 1.0f;
  v8f c = {};
  v8f d = __builtin_amdgcn_wmma_f32_16x16x4_f32(
      /*neg_a=*/false, a, /*neg_b=*/false, b,
      /*c_mod=*/(short)0, c, /*reuse_a=*/false, /*reuse_b=*/false);
  float s = d[0] + d[1] + d[2] + d[3] + d[4] + d[5] + d[6] + d[7];
  s += __shfl_xor(s, 16, 32);
  return s;   // every lane holds the full 32-lane sum
}

// ---- kernels ------------------------------------------------------------
__global__ void zero_cnt_kernel(float* cnt) {
  if (threadIdx.x < NSAMP * 3) cnt[threadIdx.x] = 0.0f;
}

__global__ void __launch_bounds__(BLK)
iou_count_kernel(const int* __restrict__ gt, const int* __restrict__ pred,
                 float* __restrict__ cnt /* [NSAMP][3] = tp,fp,fn */,
                 long long n_per_sample) {
  const long long samp = blockIdx.y;
  const int* g0 = gt   + samp * n_per_sample;
  const int* p0 = pred + samp * n_per_sample;
  const v4i* g4 = (const v4i*)g0;
  const v4i* p4 = (const v4i*)p0;
  const long long nvec = n_per_sample >> 2;

  unsigned tp = 0, fp = 0, gin_cnt = 0;

  for (long long i = (long long)blockIdx.x * BLK + threadIdx.x; i < nvec;
       i += (long long)gridDim.x * BLK) {
    v4i g = __builtin_nontemporal_load(g4 + i);   // global_load_b128, NT (stream once)
    v4i p = __builtin_nontemporal_load(p4 + i);
    acc_elem(g[0], p[0], tp, fp, gin_cnt);
    acc_elem(g[1], p[1], tp, fp, gin_cnt);
    acc_elem(g[2], p[2], tp, fp, gin_cnt);
    acc_elem(g[3], p[3], tp, fp, gin_cnt);
  }
  // scalar tail (n_per_sample not a multiple of 4) — handled by x-block 0 only
  if (blockIdx.x == 0) {
    for (long long i = (nvec << 2) + threadIdx.x; i < n_per_sample; i += BLK)
      acc_elem(g0[i], p0[i], tp, fp, gin_cnt);
  }

  const unsigned fn = gin_cnt - tp;

  // wave reduction through the matrix pipe (EXEC is all-1s here: no divergence)
  float stp = wave_sum_wmma((float)tp);
  float sfp = wave_sum_wmma((float)fp);
  float sfn = wave_sum_wmma((float)fn);

  __shared__ float red[BLK / 32][3];
  const int wave = threadIdx.x >> 5;
  const int lane = threadIdx.x & 31;
  if (lane == 0) { red[wave][0] = stp; red[wave][1] = sfp; red[wave][2] = sfn; }
  __syncthreads();
  if (threadIdx.x == 0) {
    float a = 0.f, b = 0.f, c = 0.f;
    for (int w = 0; w < BLK / 32; ++w) { a += red[w][0]; b += red[w][1]; c += red[w][2]; }
    // counts < 2^24 per sample -> float atomics are exact
    atomicAdd(&cnt[samp * 3 + 0], a);
    atomicAdd(&cnt[samp * 3 + 1], b);
    atomicAdd(&cnt[samp * 3 + 2], c);
  }
}

__global__ void finalize_kernel(const float* __restrict__ cnt, float* __restrict__ out) {
  if (threadIdx.x == 0) {
    float s = 0.0f;
    for (int b = 0; b < NSAMP; ++b) {
      float tp = cnt[3 * b + 0], fp = cnt[3 * b + 1], fn = cnt[3 * b + 2];
      float den = fmaxf(tp + fp + fn, 1e-8f);
      s += tp / den;
    }
    out[0] = s / (float)NSAMP;
  }
}

// ---- entry --------------------------------------------------------------
extern "C" void kernel_launch(void* const* d_in, const int* in_sizes, int n_in,
                              void* d_out, int out_size, void* d_ws, size_t ws_size,
                              hipStream_t stream) {
  const int* gt   = (const int*)d_in[0];
  const int* pred = (const int*)d_in[1];
  float* cnt = (float*)d_ws;                    // 24 floats of scratch
  const long long total = (long long)in_sizes[0];
  const long long nper  = total / NSAMP;

  zero_cnt_kernel<<<1, 32, 0, stream>>>(cnt);
  dim3 grid(GRIDX, NSAMP);
  iou_count_kernel<<<grid, BLK, 0, stream>>>(gt, pred, cnt, nper);
  finalize_kernel<<<1, 32, 0, stream>>>(cnt, (float*)d_out);
}